// InvConv1x1ConditionalLU_27350351741480
// MI455X (gfx1250) — compile-verified
//
#include <hip/hip_runtime.h>

#define B_    32
#define C_    192
#define H_    32
#define W_    32
#define CH_   8
#define FEAT_ 2048            // CH*16*16
#define CC_   (C_*C_)         // 36864
#define HW_   (H_*W_)         // 1024
#define OUT_ELEMS (B_*C_*H_*W_)
#define LDA_  (C_ + 1)        // 193: LU-kernel LDS stride (bank-conflict-free)
#define WPAD_ 196             // padded weight row stride; 196 % 64 == 4 -> 16 distinct banks

typedef __attribute__((ext_vector_type(2))) float v2f;
typedef __attribute__((ext_vector_type(8))) float v8f;
typedef unsigned int u32x4 __attribute__((ext_vector_type(4)));
typedef int          i32x4 __attribute__((ext_vector_type(4)));
typedef int          i32x8 __attribute__((ext_vector_type(8)));

// ---------------------------------------------------------------------------
// Flat shared pointer -> LDS byte offset (aperture keeps offset in addr[31:0]).
// ---------------------------------------------------------------------------
__device__ __forceinline__ unsigned lds_off(const void* p)
{
    return (unsigned)(unsigned long long)(uintptr_t)p;
}

// ---------------------------------------------------------------------------
// TDM: 2D tile load Global -> LDS.  data_size = 4 bytes.  Optional LDS padding
// (pad after every 2^(pad_interval+1) DWORDs, pad_amount+1 DWORDs inserted).
// D# layout per CDNA5 ISA.
// ---------------------------------------------------------------------------
__device__ __forceinline__ void tdm_load_2d_f32(unsigned lds_addr, const void* gptr,
                                                unsigned tile_d0, unsigned tile_d1,
                                                unsigned tensor_d0, unsigned tensor_d1,
                                                unsigned long long d0_stride,
                                                unsigned pad_en, unsigned pad_interval,
                                                unsigned pad_amount)
{
    const unsigned long long ga = (unsigned long long)gptr;
    u32x4 g0;
    g0[0] = 1u;                                            // count=1, user descriptor
    g0[1] = lds_addr;                                      // lds_addr [63:32]
    g0[2] = (unsigned)ga;                                  // global_addr[31:0]
    g0[3] = (unsigned)((ga >> 32) & 0x01FFFFFFu) | (2u << 30); // addr[56:32] | type=2
    i32x8 g1;
    g1[0] = (int)((2u << 16) | (pad_en << 20) |
                  (pad_interval << 22) | (pad_amount << 25)); // mask=0, data_size=4B
    g1[1] = (int)((tensor_d0 & 0xFFFFu) << 16);            // atomic_addr=0 | td0 lo16
    g1[2] = (int)(((tensor_d0 >> 16) & 0xFFFFu) | ((tensor_d1 & 0xFFFFu) << 16));
    g1[3] = (int)(((tensor_d1 >> 16) & 0xFFFFu) | ((tile_d0 & 0xFFFFu) << 16));
    g1[4] = (int)(tile_d1 & 0xFFFFu);                      // tile_dim1 | tile_dim2=0
    g1[5] = (int)(unsigned)(d0_stride & 0xFFFFFFFFull);    // dim0 stride lo32
    g1[6] = (int)(unsigned)((d0_stride >> 32) & 0xFFFFull);// dim0 stride hi16 | dim1 stride=0
    g1[7] = 0;
    i32x4 g2 = {0, 0, 0, 0};
    i32x4 g3 = {0, 0, 0, 0};
#if defined(__clang_major__) && (__clang_major__ >= 23)
    i32x8 g4 = {0, 0, 0, 0, 0, 0, 0, 0};
    __builtin_amdgcn_tensor_load_to_lds(g0, g1, g2, g3, g4, 0);
#else
    __builtin_amdgcn_tensor_load_to_lds(g0, g1, g2, g3, 0);
#endif
}

// ---------------------------------------------------------------------------
// Stage 1: Conv2d(192->8, k3, s2, p1) + ReLU + flatten -> act [32, 2048]
// ---------------------------------------------------------------------------
__global__ __launch_bounds__(256)
void k_cond_conv(const float* __restrict__ x,    // w_left_out [B,C,32,32]
                 const float* __restrict__ cw,   // [CH,C,3,3]
                 const float* __restrict__ cb,   // [CH]
                 float* __restrict__ act)        // [B, FEAT]
{
    __shared__ float ws[C_ * 9];
    const int blk = blockIdx.x;
    const int b   = blk / CH_;
    const int ch  = blk % CH_;
    const int t   = threadIdx.x;
    for (int i = t; i < C_ * 9; i += 256) ws[i] = cw[ch * C_ * 9 + i];
    __syncthreads();
    const int oy = t >> 4, ox = t & 15;
    const int iy0 = oy * 2 - 1, ix0 = ox * 2 - 1;
    float acc = cb[ch];
    const float* xb = x + (size_t)b * C_ * H_ * W_;
    for (int c = 0; c < C_; ++c) {
        const float* xc = xb + c * H_ * W_;
        const float* wc = ws + c * 9;
#pragma unroll
        for (int ky = 0; ky < 3; ++ky) {
            const int y = iy0 + ky;
            if (y < 0 || y >= H_) continue;
#pragma unroll
            for (int kx = 0; kx < 3; ++kx) {
                const int xx = ix0 + kx;
                if (xx < 0 || xx >= W_) continue;
                acc = fmaf(xc[y * W_ + xx], wc[ky * 3 + kx], acc);
            }
        }
    }
    acc = acc > 0.f ? acc : 0.f;
    act[(size_t)b * FEAT_ + ch * 256 + t] = acc;
}

// ---------------------------------------------------------------------------
// Stage 2: cond[32,36864] = act[32,2048] @ lin_w^T + lin_b
// WMMA f32 16x16x4; lin_w chunks DMA'd by the TDM into double-buffered LDS
// with TDM-inserted padding (32 DWORDs -> +1 DWORD => stride 33, conflict-free).
// ---------------------------------------------------------------------------
#define LIN_KC 32
#define LIN_NB 128
__global__ __launch_bounds__(256)
void k_cond_linear(const float* __restrict__ act,  // [32, 2048]
                   const float* __restrict__ lw,   // [36864, 2048]
                   const float* __restrict__ lb,   // [36864]
                   float* __restrict__ cond)       // [32, 36864]
{
    __shared__ float sW[2][LIN_NB][LIN_KC + 1];    // TDM fills with stride-33 padding
    __shared__ float sA[2][32][LIN_KC + 1];
    const int t    = threadIdx.x;
    const int wave = t >> 5;
    const int lane = t & 31;
    const int n0   = blockIdx.x * LIN_NB;
    const int nl   = (wave << 4) | (lane & 15);
    const int kkq  = (t & 7) << 2;                 // 0..28 step 4
    const int rq   = t >> 3;                       // 0..31
    v8f c0 = {0.f, 0.f, 0.f, 0.f, 0.f, 0.f, 0.f, 0.f};
    v8f c1 = c0;

    // preload chunk 0: TDM for weights (wave 0), cooperative for activations
    if (wave == 0)
        tdm_load_2d_f32(lds_off(&sW[0][0][0]), &lw[(size_t)n0 * FEAT_],
                        /*tile*/ LIN_KC, LIN_NB, /*tensor*/ FEAT_, CC_,
                        /*stride*/ FEAT_, /*pad*/ 1, 4, 0);
    {
        const float4 a = *(const float4*)&act[(size_t)rq * FEAT_ + kkq];
        sA[0][rq][kkq + 0] = a.x; sA[0][rq][kkq + 1] = a.y;
        sA[0][rq][kkq + 2] = a.z; sA[0][rq][kkq + 3] = a.w;
    }

    int ib = 0;
    for (int kc = 0; kc < FEAT_; kc += LIN_KC, ib ^= 1) {
        if (wave == 0) __builtin_amdgcn_s_wait_tensorcnt(0);
        __syncthreads();                            // sW[ib]/sA[ib] ready for everyone
        if (kc + LIN_KC < FEAT_) {
            if (wave == 0)                          // DMA next chunk while we compute
                tdm_load_2d_f32(lds_off(&sW[ib ^ 1][0][0]),
                                &lw[(size_t)n0 * FEAT_ + kc + LIN_KC],
                                LIN_KC, LIN_NB, FEAT_, CC_, FEAT_, 1, 4, 0);
            const float4 a = *(const float4*)&act[(size_t)rq * FEAT_ + kc + LIN_KC + kkq];
            sA[ib ^ 1][rq][kkq + 0] = a.x; sA[ib ^ 1][rq][kkq + 1] = a.y;
            sA[ib ^ 1][rq][kkq + 2] = a.z; sA[ib ^ 1][rq][kkq + 3] = a.w;
        }
        const int kh = (lane < 16) ? 0 : 2;
        const int mA = lane & 15;
#pragma unroll
        for (int k4 = 0; k4 < LIN_KC; k4 += 4) {
            v2f a0, a1, bv;
            a0.x = sA[ib][mA][k4 + kh];      a0.y = sA[ib][mA][k4 + kh + 1];
            a1.x = sA[ib][mA + 16][k4 + kh]; a1.y = sA[ib][mA + 16][k4 + kh + 1];
            bv.x = sW[ib][nl][k4 + kh];      bv.y = sW[ib][nl][k4 + kh + 1];
            c0 = __builtin_amdgcn_wmma_f32_16x16x4_f32(false, a0, false, bv,
                                                       (short)0, c0, false, false);
            c1 = __builtin_amdgcn_wmma_f32_16x16x4_f32(false, a1, false, bv,
                                                       (short)0, c1, false, false);
        }
        __syncthreads();                            // done reading buffers of this iter
    }
    // epilogue: + bias, store.  C layout: VGPR r -> row r (lanes<16) / r+8
    const float bias = lb[n0 + nl];
    const int   colg = n0 + nl;
    const int   rb   = (lane < 16) ? 0 : 8;
#pragma unroll
    for (int r = 0; r < 8; ++r) {
        const int m0 = r + rb;
        cond[(size_t)m0 * CC_ + colg]        = c0[r] + bias;
        cond[(size_t)(m0 + 16) * CC_ + colg] = c1[r] + bias;
    }
}

// ---------------------------------------------------------------------------
// Stage 3: per-batch LU (partial pivoting) fully in LDS (192x193 f32),
// then weight = P @ L @ U' (diag(U') = s_sign*exp(diag U)) into PADDED global
// layout [192][196]; also per-batch diag sum.
// ---------------------------------------------------------------------------
__global__ __launch_bounds__(256)
void k_lu_build(const float* __restrict__ cond,   // [32, 36864]
                const float* __restrict__ s_sign, // [192]
                float* __restrict__ weight,       // [32, 192, WPAD_]
                float* __restrict__ diagsum)      // [32]
{
    extern __shared__ float A[];                  // 192 x 193
    __shared__ int   perm[C_];
    __shared__ float rmax[256];
    __shared__ int   ridx[256];
    __shared__ float dexp[C_];
    __shared__ int   s_piv;
    const int b = blockIdx.x;
    const int t = threadIdx.x;
    const float* M = cond + (size_t)b * CC_;

    for (int i = t; i < CC_; i += 256) A[(i / C_) * LDA_ + (i % C_)] = M[i];
    for (int i = t; i < C_; i += 256) perm[i] = i;
    __syncthreads();

    for (int k = 0; k < C_ - 1; ++k) {
        float best = -1.f; int bi = k;
        for (int i = k + t; i < C_; i += 256) {
            const float v = fabsf(A[i * LDA_ + k]);
            if (v > best) { best = v; bi = i; }
        }
        rmax[t] = best; ridx[t] = bi;
        __syncthreads();
        if (t == 0) {
            float bb = -1.f; int pp = k;
            for (int i = 0; i < 256; ++i)
                if (rmax[i] > bb) { bb = rmax[i]; pp = ridx[i]; }
            s_piv = pp;
            if (pp != k) { const int tp = perm[k]; perm[k] = perm[pp]; perm[pp] = tp; }
        }
        __syncthreads();
        const int p = s_piv;
        if (p != k)
            for (int j = t; j < C_; j += 256) {
                const float tmp = A[k * LDA_ + j];
                A[k * LDA_ + j] = A[p * LDA_ + j];
                A[p * LDA_ + j] = tmp;
            }
        __syncthreads();
        const float inv = 1.0f / A[k * LDA_ + k];
        for (int i = k + 1 + t; i < C_; i += 256) A[i * LDA_ + k] *= inv;
        __syncthreads();
        for (int i = k + 1 + t; i < C_; i += 256) {
            const float lik = A[i * LDA_ + k];
            const float* uk = &A[k * LDA_];
            float*       ai = &A[i * LDA_];
            for (int j = k + 1; j < C_; ++j) ai[j] = fmaf(-lik, uk[j], ai[j]);
        }
        __syncthreads();
    }

    float ds = 0.f;
    for (int i = t; i < C_; i += 256) ds += A[i * LDA_ + i];
    rmax[t] = ds;
    for (int i = t; i < C_; i += 256) dexp[i] = s_sign[i] * __expf(A[i * LDA_ + i]);
    __syncthreads();
    if (t == 0) { float s = 0.f; for (int i = 0; i < 256; ++i) s += rmax[i]; diagsum[b] = s; }

    float* Wb = weight + (size_t)b * C_ * WPAD_;
    for (int e = t; e < C_ * 4; e += 256)           // zero pad columns 192..195
        Wb[(size_t)(e >> 2) * WPAD_ + C_ + (e & 3)] = 0.f;
    for (int e = t; e < CC_; e += 256) {
        const int i = e / C_, j = e % C_;
        const int mn = i < j ? i : j;
        float acc;
        if (i < j)       acc = A[i * LDA_ + j];
        else if (i == j) acc = dexp[i];
        else             acc = A[i * LDA_ + j] * dexp[j];
        for (int tt = 0; tt < mn; ++tt)
            acc = fmaf(A[i * LDA_ + tt], A[tt * LDA_ + j], acc);
        Wb[(size_t)perm[i] * WPAD_ + j] = acc;
    }
}

// ---------------------------------------------------------------------------
// Stage 4: out[b] = weight[b] (192x192) @ inp[b] (192x1024), WMMA f32 16x16x4.
// weight[b] (padded 192x196) brought into LDS by one TDM tensor load.
// ---------------------------------------------------------------------------
__global__ __launch_bounds__(256)
void k_apply(const float* __restrict__ weight,   // [32,192,WPAD_]
             const float* __restrict__ x,        // [32,192,1024]
             float* __restrict__ out)            // [32,192,1024]
{
    extern __shared__ float sWt[];               // 192 x 196
    const int b    = blockIdx.x >> 4;
    const int nb   = blockIdx.x & 15;
    const int t    = threadIdx.x;
    const int wave = t >> 5;
    const int lane = t & 31;
    const int wm   = wave & 3;                   // M-tile group {wm, wm+4, wm+8}
    const int wn   = wave >> 2;                  // N-tile group {2wn, 2wn+1}

    if (wave == 0) {
        tdm_load_2d_f32(lds_off(&sWt[0]), weight + (size_t)b * C_ * WPAD_,
                        /*tile*/ WPAD_, C_, /*tensor*/ WPAD_, C_,
                        /*stride*/ WPAD_, /*pad*/ 0, 0, 0);
        __builtin_amdgcn_s_wait_tensorcnt(0);
    }
    __syncthreads();

    const float* xb = x + (size_t)b * C_ * HW_;
    v8f acc[3][2];
#pragma unroll
    for (int mt = 0; mt < 3; ++mt)
#pragma unroll
        for (int nt = 0; nt < 2; ++nt)
            acc[mt][nt] = (v8f){0.f,0.f,0.f,0.f,0.f,0.f,0.f,0.f};

    const int kh  = (lane < 16) ? 0 : 2;
    const int l15 = lane & 15;
    for (int k0 = 0; k0 < C_; k0 += 4) {
        v2f bv[2], av[3];
#pragma unroll
        for (int nt = 0; nt < 2; ++nt) {
            const int col = nb * 64 + (wn * 2 + nt) * 16 + l15;
            bv[nt].x = xb[(size_t)(k0 + kh)     * HW_ + col];
            bv[nt].y = xb[(size_t)(k0 + kh + 1) * HW_ + col];
        }
#pragma unroll
        for (int mt = 0; mt < 3; ++mt) {
            const int m = (wm + mt * 4) * 16 + l15;
            av[mt].x = sWt[m * WPAD_ + k0 + kh];
            av[mt].y = sWt[m * WPAD_ + k0 + kh + 1];
        }
#pragma unroll
        for (int mt = 0; mt < 3; ++mt)
#pragma unroll
            for (int nt = 0; nt < 2; ++nt)
                acc[mt][nt] = __builtin_amdgcn_wmma_f32_16x16x4_f32(
                    false, av[mt], false, bv[nt], (short)0, acc[mt][nt], false, false);
    }

    float* ob = out + (size_t)b * C_ * HW_;
    const int rb = (lane < 16) ? 0 : 8;
#pragma unroll
    for (int mt = 0; mt < 3; ++mt)
#pragma unroll
        for (int nt = 0; nt < 2; ++nt) {
            const int col = nb * 64 + (wn * 2 + nt) * 16 + l15;
#pragma unroll
            for (int r = 0; r < 8; ++r) {
                const int m = (wm + mt * 4) * 16 + r + rb;
                ob[(size_t)m * HW_ + col] = acc[mt][nt][r];
            }
        }
}

// ---------------------------------------------------------------------------
// Stage 5: log_det = (H*W) * mean_b( sum diag(U) ) -> d_out[OUT_ELEMS]
// ---------------------------------------------------------------------------
__global__ void k_logdet(const float* __restrict__ diagsum, float* __restrict__ outv)
{
    if (threadIdx.x == 0) {
        float s = 0.f;
        for (int i = 0; i < B_; ++i) s += diagsum[i];
        outv[0] = (float)HW_ * (s / (float)B_);
    }
}

// ---------------------------------------------------------------------------
extern "C" void kernel_launch(void* const* d_in, const int* in_sizes, int n_in,
                              void* d_out, int out_size, void* d_ws, size_t ws_size,
                              hipStream_t stream)
{
    (void)in_sizes; (void)n_in; (void)out_size; (void)ws_size;
    const float* inp    = (const float*)d_in[0];
    const float* wlo    = (const float*)d_in[1];
    const float* conv_w = (const float*)d_in[2];
    const float* conv_b = (const float*)d_in[3];
    const float* lin_w  = (const float*)d_in[4];
    const float* lin_b  = (const float*)d_in[5];
    const float* s_sign = (const float*)d_in[6];
    float* out = (float*)d_out;

    float* ws     = (float*)d_ws;
    float* act    = ws;                             // 32*2048
    float* cond   = act  + (size_t)B_ * FEAT_;      // 32*36864
    float* wmat   = cond + (size_t)B_ * CC_;        // 32*192*196 (padded)
    float* dsum   = wmat + (size_t)B_ * C_ * WPAD_; // 32

    const size_t lds_lu    = (size_t)C_ * LDA_  * sizeof(float);  // 148224 B
    const size_t lds_apply = (size_t)C_ * WPAD_ * sizeof(float);  // 150528 B

    k_cond_conv  <<<B_ * CH_,     256, 0,         stream>>>(wlo, conv_w, conv_b, act);
    k_cond_linear<<<CC_ / LIN_NB, 256, 0,         stream>>>(act, lin_w, lin_b, cond);
    k_lu_build   <<<B_,           256, lds_lu,    stream>>>(cond, s_sign, wmat, dsum);
    k_apply      <<<B_ * 16,      256, lds_apply, stream>>>(wmat, inp, out);
    k_logdet     <<<1,            32,  0,         stream>>>(dsum, out + (size_t)OUT_ELEMS);
}